// GatherBlock_41420664602704
// MI455X (gfx1250) — compile-verified
//
#include <hip/hip_runtime.h>
#include <stdint.h>

// ---------------------------------------------------------------------------
// GatherBlock on MI455X (gfx1250) via the Tensor Data Mover (TDM).
//
// Problem constants (from the reference):
//   x:  (N=8, C=64, H=256, W=256) fp32
//   idx:(M=1024, 3) int32  [b, yb, xb]
//   out:(M, C, 16, 16) fp32
//
// Each tile = 64 KiB. TDM 3D descriptor: tile_dim = (BW=16, BH=16, C=64),
// tensor strides = (W, H*W) elements. TDM fills LDS in X->Y->Z order =
// (x, y, c) fastest-to-slowest = exactly the (c, y, x) output layout, so a
// 1D tensor_store_from_lds writes the tile contiguously to the output.
// ---------------------------------------------------------------------------

#define NN 8
#define CC 64
#define HH 256
#define WW 256
#define BH 16
#define BW 16
#define TILE_ELEMS (CC * BH * BW)      // 16384 fp32
#define TILE_BYTES (TILE_ELEMS * 4)    // 65536 bytes
#define TILES_PER_BLOCK 4

typedef __attribute__((ext_vector_type(4))) unsigned int u32x4;
typedef __attribute__((ext_vector_type(8))) unsigned int u32x8;

// D# group 0 (128b): [1:0]=count, [63:32]=lds_addr, [120:64]=global_addr,
// [127:126]=type(2).
__device__ __forceinline__ u32x4 tdm_group0(uint32_t lds_byte_off, uint64_t gaddr) {
  u32x4 g;
  g.x = 1u;                                                   // count=1, user mode
  g.y = lds_byte_off;                                         // lds_addr
  g.z = (uint32_t)(gaddr & 0xFFFFFFFFu);                      // global_addr[31:0]
  g.w = (uint32_t)((gaddr >> 32) & 0x01FFFFFFu) | (2u << 30); // addr[56:32] | type=2
  return g;
}

// D# group 1 (256b): data_size[17:16], tensor_dim0[79:48], tensor_dim1[111:80],
// tile_dim0[127:112], tile_dim1[143:128], tile_dim2[159:144],
// tensor_dim0_stride[207:160], tensor_dim1_stride[255:208].
__device__ __forceinline__ u32x8 tdm_group1(uint32_t tdim0, uint32_t tdim1,
                                            uint32_t tile0, uint32_t tile1,
                                            uint32_t tile2,
                                            uint64_t stride0, uint64_t stride1) {
  u32x8 g;
  g.s0 = 2u << 16;                                            // data_size = 4 bytes
  g.s1 = (tdim0 & 0xFFFFu) << 16;                             // tensor_dim0[15:0]
  g.s2 = (tdim0 >> 16) | ((tdim1 & 0xFFFFu) << 16);           // td0[31:16] | td1[15:0]
  g.s3 = ((tdim1 >> 16) & 0xFFFFu) | (tile0 << 16);           // td1[31:16] | tile_dim0
  g.s4 = (tile1 & 0xFFFFu) | (tile2 << 16);                   // tile_dim1 | tile_dim2
  g.s5 = (uint32_t)(stride0 & 0xFFFFFFFFu);                   // stride0[31:0]
  g.s6 = (uint32_t)((stride0 >> 32) & 0xFFFFu) |
         ((uint32_t)(stride1 & 0xFFFFu) << 16);               // stride0[47:32]|stride1[15:0]
  g.s7 = (uint32_t)((stride1 >> 16) & 0xFFFFFFFFu);           // stride1[47:16]
  return g;
}

__global__ __launch_bounds__(32)
void GatherBlock_tdm_kernel(const float* __restrict__ x,
                            const int* __restrict__ idx,
                            float* __restrict__ out,
                            int M) {
  // Two 64 KiB LDS tile buffers (double buffer) = 128 KiB of the 320 KiB WGP LDS.
  __shared__ float lds_tile[2][TILE_ELEMS];

  const int m0 = blockIdx.x * TILES_PER_BLOCK;

  const uint64_t xbase   = (uint64_t)(uintptr_t)x;
  const uint64_t outbase = (uint64_t)(uintptr_t)out;

  // Group2 for the 3D load: tensor_dim2 = C (bits[31:0]); rest unused/zero.
  const u32x4 g2_load = {(uint32_t)CC, 0u, 0u, 0u};
  const u32x4 g_zero  = {0u, 0u, 0u, 0u};

  // Shared tensor-shape descriptor pieces (per-tile only global_addr changes).
  // Load: 3D tile (16 x 16 x 64), strides in elements: dim0=W, dim1=H*W.
  const u32x8 g1_load = tdm_group1(/*tdim0=*/BW, /*tdim1=*/BH,
                                   /*tile0=*/BW, /*tile1=*/BH, /*tile2=*/CC,
                                   /*stride0=*/(uint64_t)WW,
                                   /*stride1=*/(uint64_t)HH * WW);
  // Store: 1D tile of 16384 contiguous elements.
  const u32x8 g1_store = tdm_group1(/*tdim0=*/TILE_ELEMS, /*tdim1=*/0,
                                    /*tile0=*/TILE_ELEMS, /*tile1=*/0, /*tile2=*/0,
                                    /*stride0=*/(uint64_t)TILE_ELEMS,
                                    /*stride1=*/0);

  auto issue_load = [&](int t, int buf) {
    if (t >= M) return;
    // Uniform per-block scalar index fetch.
    const int b  = __builtin_amdgcn_readfirstlane(idx[3 * t + 0]);
    const int yb = __builtin_amdgcn_readfirstlane(idx[3 * t + 1]);
    const int xb = __builtin_amdgcn_readfirstlane(idx[3 * t + 2]);
    const uint64_t elem_off = (uint64_t)b * ((uint64_t)CC * HH * WW) +
                              (uint64_t)yb * ((uint64_t)BH * WW) +
                              (uint64_t)xb * (uint64_t)BW;
    const uint64_t gaddr = xbase + elem_off * 4ull;
    const uint32_t lds_off = (uint32_t)(uintptr_t)(&lds_tile[buf][0]);
    const u32x4 g0 = tdm_group0(lds_off, gaddr);
    asm volatile("tensor_load_to_lds %0, %1, %2, %3"
                 :
                 : "s"(g0), "s"(g1_load), "s"(g2_load), "s"(g_zero)
                 : "memory");
  };

  auto issue_store = [&](int t, int buf) {
    if (t >= M) return;
    const uint64_t gaddr = outbase + (uint64_t)t * (uint64_t)TILE_BYTES;
    const uint32_t lds_off = (uint32_t)(uintptr_t)(&lds_tile[buf][0]);
    const u32x4 g0 = tdm_group0(lds_off, gaddr);
    asm volatile("tensor_store_from_lds %0, %1, %2, %3"
                 :
                 : "s"(g0), "s"(g1_store), "s"(g_zero), "s"(g_zero)
                 : "memory");
  };

  // Software pipeline: TDM ops from one wave execute & complete in order, so
  // after issuing load(i+1), waiting TENSORcnt<=1 guarantees load(i) (and all
  // earlier stores) finished; store(i) then reads a complete LDS tile while
  // load(i+1) streams into the other buffer.
  issue_load(m0 + 0, 0);
#pragma unroll
  for (int i = 0; i < TILES_PER_BLOCK; ++i) {
    if (i + 1 < TILES_PER_BLOCK) {
      issue_load(m0 + i + 1, (i + 1) & 1);
      __builtin_amdgcn_s_wait_tensorcnt(1);
    } else {
      __builtin_amdgcn_s_wait_tensorcnt(0);
    }
    issue_store(m0 + i, i & 1);
  }
  __builtin_amdgcn_s_wait_tensorcnt(0);
}

extern "C" void kernel_launch(void* const* d_in, const int* in_sizes, int n_in,
                              void* d_out, int out_size, void* d_ws, size_t ws_size,
                              hipStream_t stream) {
  const float* x   = (const float*)d_in[0];
  const int*   idx = (const int*)d_in[1];
  float*       out = (float*)d_out;

  const int M = in_sizes[1] / 3;  // active_indices is (M, 3)
  const int blocks = (M + TILES_PER_BLOCK - 1) / TILES_PER_BLOCK;

  GatherBlock_tdm_kernel<<<dim3(blocks), dim3(32), 0, stream>>>(x, idx, out, M);
}